// SoftDepthShader_85796266705111
// MI455X (gfx1250) — compile-verified
//
#include <hip/hip_runtime.h>

// SoftDepthShader (pytorch3d softmax_rgb_blend, blue channel) for MI455X.
//
// Memory-bound streaming reduction: one wave32 per pixel, lanes own K-pairs.
// Non-temporal b64 loads (stream 400+ MB, don't thrash 192MB L2), wave32
// shuffle reductions (ds_bpermute), NT store of the single output per pixel.

typedef float v2f __attribute__((ext_vector_type(2)));
typedef int   v2i __attribute__((ext_vector_type(2)));

namespace {
constexpr float kSigma = 1e-4f;
constexpr float kGamma = 1e-4f;
constexpr float kZNear = 1.0f;
constexpr float kZFar  = 100.0f;
constexpr float kEps   = 1e-10f;
constexpr int   kK     = 50;          // slots per pixel
constexpr int   kWavesPerBlock = 8;   // 256 threads = 8 wave32
}

__device__ __forceinline__ float wave_max32(float v) {
#pragma unroll
  for (int off = 16; off > 0; off >>= 1)
    v = fmaxf(v, __shfl_xor(v, off, 32));
  return v;
}

__device__ __forceinline__ float wave_sum32(float v) {
#pragma unroll
  for (int off = 16; off > 0; off >>= 1)
    v += __shfl_xor(v, off, 32);
  return v;
}

__global__ __launch_bounds__(kWavesPerBlock * 32)
void soft_depth_blend_kernel(const float* __restrict__ zbuf,
                             const float* __restrict__ dists,
                             const int*   __restrict__ face,   // int32; sign test == int64 sign test
                             float* __restrict__ out,
                             int npix)
{
  const int lane = threadIdx.x & 31;
  const int pix  = blockIdx.x * kWavesPerBlock + (threadIdx.x >> 5);
  if (pix >= npix) return;

  // Element base of this pixel's K-span; pix*50 is even -> 8B-aligned v2 loads.
  const long long base = (long long)pix * kK;
  const int k0 = lane << 1;  // lane's first K index (0..62)

  float z0 = 0.f, z1 = 0.f, d0 = 0.f, d1 = 0.f;
  int   f0 = -1,  f1 = -1;
  if (k0 < kK) {  // lanes 0..24 load; 25..31 stay masked (no OOB at buffer tail)
    v2f zz = __builtin_nontemporal_load((const v2f*)(zbuf  + base) + lane);
    v2f dd = __builtin_nontemporal_load((const v2f*)(dists + base) + lane);
    v2i ff = __builtin_nontemporal_load((const v2i*)(face  + base) + lane);
    z0 = zz.x; z1 = zz.y;
    d0 = dd.x; d1 = dd.y;
    f0 = ff.x; f1 = ff.y;
  }
  const bool m0 = (f0 >= 0) && (k0     < kK);
  const bool m1 = (f1 >= 0) && (k0 + 1 < kK);

  // Masked normalized inverse depth (reference: z_inv = (zfar-z)/(zfar-znear)*mask)
  constexpr float inv_range = 1.0f / (kZFar - kZNear);
  const float zi0 = m0 ? (kZFar - z0) * inv_range : 0.f;
  const float zi1 = m1 ? (kZFar - z1) * inv_range : 0.f;

  // Per-pixel softmax stabilizer: max over all K (masked entries contribute 0,
  // matching the reference), clamped to EPS.
  const float zmax = fmaxf(wave_max32(fmaxf(zi0, zi1)), kEps);

  // Distance sigmoid: sigmoid(-d/sigma) = 1/(1+exp(d/sigma)); masked -> 0.
  constexpr float inv_sigma = 1.0f / kSigma;
  const float p0 = m0 ? 1.0f / (1.0f + expf(d0 * inv_sigma)) : 0.f;
  const float p1 = m1 ? 1.0f / (1.0f + expf(d1 * inv_sigma)) : 0.f;

  // Softmax numerators; exponent args are <= 0, no overflow possible.
  constexpr float inv_gamma = 1.0f / kGamma;
  const float w0 = p0 * expf((zi0 - zmax) * inv_gamma);
  const float w1 = p1 * expf((zi1 - zmax) * inv_gamma);

  const float sum_w  = wave_sum32(w0 + w1);
  const float sum_wz = wave_sum32(w0 * z0 + w1 * z1);

  if (lane == 0) {
    const float delta = fmaxf(expf((kEps - zmax) * inv_gamma), kEps);
    // BG_BLUE == 1.0 -> numerator gets +delta*1
    const float blue = (sum_wz + delta) / (sum_w + delta);
    __builtin_nontemporal_store(blue, out + pix);
  }
}

extern "C" void kernel_launch(void* const* d_in, const int* in_sizes, int n_in,
                              void* d_out, int out_size, void* d_ws, size_t ws_size,
                              hipStream_t stream) {
  (void)n_in; (void)d_ws; (void)ws_size; (void)in_sizes;
  const float* zbuf  = (const float*)d_in[0];
  const float* dists = (const float*)d_in[1];
  const int*   face  = (const int*)d_in[2];
  float*       out   = (float*)d_out;

  const int npix   = out_size;  // N*H*W; each output pixel reduces K=50 slots
  const int blocks = (npix + kWavesPerBlock - 1) / kWavesPerBlock;
  soft_depth_blend_kernel<<<blocks, kWavesPerBlock * 32, 0, stream>>>(
      zbuf, dists, face, out, npix);
}